// DistanceCellList_38534446580205
// MI455X (gfx1250) — compile-verified
//
#include <hip/hip_runtime.h>

// Neighbor-pair search (TorchMD DistanceCellList) for gfx1250 / MI455X.
//
// d2[i][j] = |pi|^2 - 2 pi.pj + |pj|^2 is a rank-4 GEMM -> V_WMMA_F32_16X16X4_F32
// (A[m]=(x,y,z,|p|^2), B[:,n]=(-2x,-2y,-2z,1), C[m][n]=|pj|^2). One WMMA = 256 pair
// distances; the whole 4096x4096 upper triangle is ~33K WMMAs. Inputs (48KB pos)
// live in L2; the problem is matrix-pipe + compaction bound, HBM irrelevant.
//
// Pass 1: WMMA tile sweep -> per-row valid counts (wave-uniform scalar popcounts).
// Pass 2: exclusive prefix sum over rows -> lex-ordered (i-major) slot bases.
// Pass 3: identical WMMA predicate, ballot-ranked ordered writes; running per-row
//         offsets kept as wave-uniform registers (no LDS, no barriers in hot loop).

#define MAX_PAIRS 32768
#define CUTOFF2   25.0f   // 5.0^2
#define TILE      16

typedef __attribute__((ext_vector_type(2))) float v2f;
typedef __attribute__((ext_vector_type(8))) float v8f;

// Process one 16x16 tile: compute d2 via WMMA, build per-row 16-bit validity masks
// (wave-uniform), hand them to the caller. DIAG applies the i<j triangular constraint
// as compile-time constant mask ANDs (zero VALU cost).
template <bool DIAG, typename PerRow>
__device__ __forceinline__ void process_tile(const float* __restrict__ pos,
                                             const int* __restrict__ batch,
                                             v2f A, const int (&vrb)[8],
                                             int j0, int lane, PerRow&& perRow) {
  const int  n  = lane & 15;
  const bool hi = lane >= 16;
  const int  j  = j0 + n;

  const float bx = pos[3 * j + 0];
  const float by = pos[3 * j + 1];
  const float bz = pos[3 * j + 2];
  const float bn2 = bx * bx + by * by + bz * bz;
  const int   bb  = batch[j];

  // B operand: lanes 0-15 carry (K0,K1)=(-2x,-2y); lanes 16-31 carry (K2,K3)=(-2z,1)
  v2f B;
  B.x = hi ? (-2.0f * bz) : (-2.0f * bx);
  B.y = hi ? 1.0f         : (-2.0f * by);
  v8f C;
#pragma unroll
  for (int v = 0; v < 8; ++v) C[v] = bn2;

  v8f D = __builtin_amdgcn_wmma_f32_16x16x4_f32(
      false, A, false, B, (short)0, C, false, false);

#pragma unroll
  for (int v = 0; v < 8; ++v) {
    // lane covers row (v | v+8) depending on half; vrb[v] already half-selected
    bool pre = (D[v] < CUTOFF2) && (vrb[v] == bb);
    unsigned long long m = __ballot(pre ? 1 : 0);
    unsigned mlo = (unsigned)m & 0xFFFFu;          // row v      (cols 0..15)
    unsigned mhi = (unsigned)(m >> 16) & 0xFFFFu;  // row v + 8  (cols 0..15)
    if (DIAG) {
      mlo &= (0xFFFFu << (v + 1)) & 0xFFFFu;                         // keep n > v
      mhi &= (v + 9 < 16) ? ((0xFFFFu << (v + 9)) & 0xFFFFu) : 0u;   // keep n > v+8
    }
    perRow(v, mlo, mhi, bx, by, bz, j);
  }
}

// ---------------- Pass 0: initialize outputs ----------------
__global__ __launch_bounds__(256) void init_out_kernel(float* out, int total) {
  int idx = blockIdx.x * blockDim.x + threadIdx.x;
  if (idx >= total) return;
  out[idx] = (idx < 2 * MAX_PAIRS) ? -1.0f : 0.0f;  // neighbors = -1, dist/vec = 0
}

// ---------------- Pass 1: per-row valid-pair counts ----------------
__global__ __launch_bounds__(128) void count_kernel(const float* __restrict__ pos,
                                                    const int* __restrict__ batch,
                                                    int* __restrict__ rowCount,
                                                    int ntile) {
  const int rt   = blockIdx.x;
  const int i0   = rt * TILE;
  const int lane = threadIdx.x & 31;
  const int wave = threadIdx.x >> 5;  // 0..3

  __shared__ int rcnt[TILE];
  if (threadIdx.x < TILE) rcnt[threadIdx.x] = 0;
  __syncthreads();

  const int  n  = lane & 15;
  const bool hi = lane >= 16;

  // A operand: lanes 0-15 carry (K0,K1)=(x,y); lanes 16-31 carry (K2,K3)=(z,|p|^2)
  const int ia = i0 + n;
  const float ax = pos[3 * ia + 0];
  const float ay = pos[3 * ia + 1];
  const float az = pos[3 * ia + 2];
  const float an2 = ax * ax + ay * ay + az * az;
  v2f A;
  A.x = hi ? az : ax;
  A.y = hi ? an2 : ay;

  int vrb[8];  // batch id of the row this (lane-half, v) covers
#pragma unroll
  for (int v = 0; v < 8; ++v) vrb[v] = batch[i0 + v + (hi ? 8 : 0)];

  int cLo[8] = {0, 0, 0, 0, 0, 0, 0, 0};
  int cHi[8] = {0, 0, 0, 0, 0, 0, 0, 0};
  auto accum = [&](int v, unsigned mlo, unsigned mhi, float, float, float, int) {
    cLo[v] += __popc(mlo);   // wave-uniform scalar math
    cHi[v] += __popc(mhi);
  };

  int ct = rt + wave;
  if (wave == 0) {  // wave-uniform branch: diagonal tile
    process_tile<true>(pos, batch, A, vrb, ct * TILE, lane, accum);
    ct += 4;
  }
  for (; ct < ntile; ct += 4)
    process_tile<false>(pos, batch, A, vrb, ct * TILE, lane, accum);

  if (lane == 0) {  // counts are uniform; one lane per wave commits them
#pragma unroll
    for (int v = 0; v < 8; ++v) {
      atomicAdd(&rcnt[v], cLo[v]);
      atomicAdd(&rcnt[v + 8], cHi[v]);
    }
  }
  __syncthreads();
  if (threadIdx.x < TILE) rowCount[i0 + threadIdx.x] = rcnt[threadIdx.x];
}

// ---------------- Pass 2: exclusive prefix sum over rows ----------------
__global__ __launch_bounds__(256) void scan_kernel(const int* __restrict__ rowCount,
                                                   int* __restrict__ rowStart,
                                                   int nrows) {
  __shared__ int part[256];
  int t = threadIdx.x;
  int chunk = nrows / 256;  // 16
  int base = t * chunk;
  int s = 0;
  for (int k = 0; k < chunk; ++k) s += rowCount[base + k];
  part[t] = s;
  __syncthreads();
  for (int off = 1; off < 256; off <<= 1) {
    int v = (t >= off) ? part[t - off] : 0;
    __syncthreads();
    part[t] += v;
    __syncthreads();
  }
  int start = part[t] - s;  // exclusive start of this chunk
  for (int k = 0; k < chunk; ++k) {
    rowStart[base + k] = start;
    start += rowCount[base + k];
  }
}

// ---------------- Pass 3: ranked, ordered writes ----------------
__global__ __launch_bounds__(32) void write_kernel(const float* __restrict__ pos,
                                                   const int* __restrict__ batch,
                                                   const int* __restrict__ rowStart,
                                                   float* __restrict__ nb0,
                                                   float* __restrict__ nb1,
                                                   float* __restrict__ dist,
                                                   float* __restrict__ vecs,
                                                   int ntile) {
  const int rt   = blockIdx.x;
  const int i0   = rt * TILE;
  const int lane = threadIdx.x;  // one wave per block
  const int  n   = lane & 15;
  const bool hi  = lane >= 16;

  const int ia = i0 + n;
  const float ax = pos[3 * ia + 0];
  const float ay = pos[3 * ia + 1];
  const float az = pos[3 * ia + 2];
  const float an2 = ax * ax + ay * ay + az * az;
  v2f A;
  A.x = hi ? az : ax;
  A.y = hi ? an2 : ay;

  int   vrb[8];
  float wx[8], wy[8], wz[8];  // row positions for exact vec/dist values
#pragma unroll
  for (int v = 0; v < 8; ++v) {
    int r = i0 + v + (hi ? 8 : 0);
    vrb[v] = batch[r];
    wx[v] = pos[3 * r + 0];
    wy[v] = pos[3 * r + 1];
    wz[v] = pos[3 * r + 2];
  }

  // running slot offset per row: wave-uniform, fully register-resident
  int roff[16];
#pragma unroll
  for (int k = 0; k < 16; ++k) roff[k] = rowStart[i0 + k];

  auto writer = [&](int v, unsigned mlo, unsigned mhi,
                    float bx, float by, float bz, int j) {
    unsigned half = hi ? mhi : mlo;
    int rank = __popc(half & ((1u << n) - 1u));  // rank within this row's segment
    int base = hi ? roff[v + 8] : roff[v];
    int slot = base + rank;
    if (((half >> n) & 1u) && slot < MAX_PAIRS) {
      int gi = i0 + v + (hi ? 8 : 0);
      // exact values, matching reference arithmetic: vec = pos[i] - pos[j]
      float dx = wx[v] - bx;
      float dy = wy[v] - by;
      float dz = wz[v] - bz;
      nb0[slot]  = (float)gi;
      nb1[slot]  = (float)j;
      dist[slot] = sqrtf(dx * dx + dy * dy + dz * dz);
      vecs[3 * slot + 0] = dx;
      vecs[3 * slot + 1] = dy;
      vecs[3 * slot + 2] = dz;
    }
    roff[v]     += __popc(mlo);  // uniform scalar bookkeeping
    roff[v + 8] += __popc(mhi);
  };

  process_tile<true>(pos, batch, A, vrb, rt * TILE, lane, writer);  // diagonal
  for (int ct = rt + 1; ct < ntile; ++ct)
    process_tile<false>(pos, batch, A, vrb, ct * TILE, lane, writer);
}

extern "C" void kernel_launch(void* const* d_in, const int* in_sizes, int n_in,
                              void* d_out, int out_size, void* d_ws, size_t ws_size,
                              hipStream_t stream) {
  const float* pos   = (const float*)d_in[0];
  const int*   batch = (const int*)d_in[1];
  const int N     = in_sizes[1];  // 4096 atoms
  const int ntile = N / TILE;     // 256

  float* out  = (float*)d_out;
  float* nb0  = out;                  // neighbors row 0 (ii)
  float* nb1  = out + MAX_PAIRS;      // neighbors row 1 (jj)
  float* dist = out + 2 * MAX_PAIRS;  // distances
  float* vecs = out + 3 * MAX_PAIRS;  // distance_vecs [MAX_PAIRS,3]

  int* rowCount = (int*)d_ws;      // N ints
  int* rowStart = rowCount + N;    // N ints

  const int total = 6 * MAX_PAIRS;  // 196608 output floats
  init_out_kernel<<<(total + 255) / 256, 256, 0, stream>>>(out, total);
  count_kernel<<<ntile, 128, 0, stream>>>(pos, batch, rowCount, ntile);
  scan_kernel<<<1, 256, 0, stream>>>(rowCount, rowStart, N);
  write_kernel<<<ntile, 32, 0, stream>>>(pos, batch, rowStart, nb0, nb1, dist, vecs, ntile);
}